// NoEmbed_AssociativeNet_40475771797932
// MI455X (gfx1250) — compile-verified
//
#include <hip/hip_runtime.h>
#include <math.h>

// Problem dims (fixed by the reference)
#define B_ 128
#define T_ 32
#define D_ 128
#define H_ 256
#define O_ 128
#define Z_ (D_ + H_)   // 384

typedef __attribute__((ext_vector_type(2))) float v2f;
typedef __attribute__((ext_vector_type(8))) float v8f;

// ---------------------------------------------------------------------------
// Kernel 1: sequential STPN recurrence, one workgroup (256 thr = 8 waves) per
// batch element. F_b (256x384 fp32 = 384 KB) lives IN PLACE in the d_out F
// region; total F state (50.3 MB) is L2-resident (192 MB L2), so the step
// loop streams from L2, never HBM.
//
// Fused single pass over F per timestep: each wave owns 32 rows; for a row,
// the 32 lanes cooperatively load 12-element chunks of W/F/lam/gamma
// (coalesced 1.5 KB per row per array), compute the partial dot with z (LDS),
// butterfly-reduce so all lanes hold pre, take tanh, and update the F chunk
// still held in registers before storing it. F is read ONCE and written once
// per step (vs read twice in a two-pass scheme) — 33% less traffic on the
// binding resource.
// ---------------------------------------------------------------------------
__global__ __launch_bounds__(256) void stpn_recurrent_kernel(
    const float* __restrict__ sentence,  // [B,T,D]
    const float* __restrict__ h0,        // [B,H]
    const float* __restrict__ F0,        // [B,H,Z]
    const float* __restrict__ W,         // [H,Z]
    const float* __restrict__ bias,      // [H]
    const float* __restrict__ lam,       // [H,Z]
    const float* __restrict__ gam,       // [H,Z]
    float* __restrict__ outH,            // [B,H]
    float* __restrict__ outF)            // [B,H,Z], also working state
{
  const int b    = blockIdx.x;
  const int tid  = threadIdx.x;
  const int w    = tid >> 5;             // wave id 0..7
  const int lane = tid & 31;

  __shared__ __align__(16) float z[Z_];  // [x_t ; h_{t-1}]
  __shared__ float hbuf[H_];

  float*       Fb  = outF + (size_t)b * H_ * Z_;
  const float* F0b = F0   + (size_t)b * H_ * Z_;

  // Initialize working F from F0 and h from h0 (fully rewritten every call).
  {
    const float4* src = (const float4*)F0b;
    float4*       dst = (float4*)Fb;
    const int n4 = (H_ * Z_) / 4;        // 24576 float4s
    for (int i = tid; i < n4; i += 256) dst[i] = src[i];
    hbuf[tid] = h0[b * H_ + tid];
  }
  __syncthreads();

  const float* xb = sentence + (size_t)b * T_ * D_;
  // float4-chunk index of this lane's 12-element slice of a row
  const int c0 = lane * 3;               // 3 float4 = 12 floats per lane

  for (int t = 0; t < T_; ++t) {
    // Build z = [x_t ; h] in LDS
    if (tid < D_) z[tid] = xb[t * D_ + tid];
    z[D_ + tid] = hbuf[tid];
    // Prefetch next timestep's x row while we chew on F (global_prefetch_b8)
    if (t + 1 < T_ && tid < D_ / 16)
      __builtin_prefetch(xb + (t + 1) * D_ + tid * 16, 0, 0);
    __syncthreads();

    // This lane's z slice is row-invariant: hoist out of the row loop.
    const float4* z4p = (const float4*)z;
    const float4 zc0 = z4p[c0 + 0];
    const float4 zc1 = z4p[c0 + 1];
    const float4 zc2 = z4p[c0 + 2];

    for (int rr = 0; rr < 32; ++rr) {
      const int r = w * 32 + rr;
      const size_t rowOff = (size_t)r * (Z_ / 4) + c0;
      const float4* Wr = (const float4*)W   + rowOff;
      float4*       Fr = (float4*)Fb        + rowOff;
      const float4* Lr = (const float4*)lam + rowOff;
      const float4* Gr = (const float4*)gam + rowOff;

      const float4 w0 = Wr[0], w1 = Wr[1], w2 = Wr[2];
      float4 f0 = Fr[0], f1 = Fr[1], f2 = Fr[2];

      // Partial dot over this lane's 12 elements of (W+F) . z
      float p = 0.f;
      p = fmaf(w0.x + f0.x, zc0.x, p); p = fmaf(w0.y + f0.y, zc0.y, p);
      p = fmaf(w0.z + f0.z, zc0.z, p); p = fmaf(w0.w + f0.w, zc0.w, p);
      p = fmaf(w1.x + f1.x, zc1.x, p); p = fmaf(w1.y + f1.y, zc1.y, p);
      p = fmaf(w1.z + f1.z, zc1.z, p); p = fmaf(w1.w + f1.w, zc1.w, p);
      p = fmaf(w2.x + f2.x, zc2.x, p); p = fmaf(w2.y + f2.y, zc2.y, p);
      p = fmaf(w2.z + f2.z, zc2.z, p); p = fmaf(w2.w + f2.w, zc2.w, p);

      // Butterfly reduce across the wave32 — every lane ends with the total.
      #pragma unroll
      for (int m = 16; m >= 1; m >>= 1) p += __shfl_xor(p, m, 32);

      const float hn = tanhf(p + bias[r]);
      if (lane == 0) hbuf[r] = hn;       // old h already captured in z

      // Hebbian update of the SAME chunk, still in registers:
      // F = lam*F + gamma*hn*z
      const float4 l0 = Lr[0], l1 = Lr[1], l2 = Lr[2];
      const float4 g0 = Gr[0], g1 = Gr[1], g2 = Gr[2];
      f0.x = fmaf(l0.x, f0.x, g0.x * hn * zc0.x);
      f0.y = fmaf(l0.y, f0.y, g0.y * hn * zc0.y);
      f0.z = fmaf(l0.z, f0.z, g0.z * hn * zc0.z);
      f0.w = fmaf(l0.w, f0.w, g0.w * hn * zc0.w);
      f1.x = fmaf(l1.x, f1.x, g1.x * hn * zc1.x);
      f1.y = fmaf(l1.y, f1.y, g1.y * hn * zc1.y);
      f1.z = fmaf(l1.z, f1.z, g1.z * hn * zc1.z);
      f1.w = fmaf(l1.w, f1.w, g1.w * hn * zc1.w);
      f2.x = fmaf(l2.x, f2.x, g2.x * hn * zc2.x);
      f2.y = fmaf(l2.y, f2.y, g2.y * hn * zc2.y);
      f2.z = fmaf(l2.z, f2.z, g2.z * hn * zc2.z);
      f2.w = fmaf(l2.w, f2.w, g2.w * hn * zc2.w);
      Fr[0] = f0; Fr[1] = f1; Fr[2] = f2;
    }
    __syncthreads();                     // before next timestep rebuilds z
  }

  outH[b * H_ + tid] = hbuf[tid];        // final hidden state -> output
}

// ---------------------------------------------------------------------------
// Kernel 2: head GEMM tag = h @ W_out^T + b_out  ([128,256] x [256,128]).
// fp32 WMMA path: V_WMMA_F32_16X16X4_F32, one wave per 16x16 output tile,
// K looped 256/4 = 64 times. Layouts per ISA 7.12.2 (32-bit A 16x4, 32-bit
// C/D 16x16): lanes 0-15 take K (resp. M) low half, lanes 16-31 high half.
// EXEC is all-ones (exactly 32 threads per block, no divergence).
// ---------------------------------------------------------------------------
__global__ __launch_bounds__(32) void head_wmma_kernel(
    const float* __restrict__ hmat,   // [B,H]  (A: M=B, K=H)
    const float* __restrict__ Wout,   // [O,H]  (B^T: Wout[n*H + k] = B[k][n])
    const float* __restrict__ bout,   // [O]
    float* __restrict__ tag)          // [B,O]
{
  const int tileM = blockIdx.x;        // 0..7  (B/16)
  const int tileN = blockIdx.y;        // 0..7  (O/16)
  const int lane  = threadIdx.x;       // wave32
  const int half  = lane >> 4;         // 0: lanes 0-15, 1: lanes 16-31
  const int l16   = lane & 15;

  const int m = tileM * 16 + l16;      // A row owned by this lane
  const int n = tileN * 16 + l16;      // B column owned by this lane

  v8f c = {};
  for (int k0 = 0; k0 < H_; k0 += 4) {
    const int ka = k0 + 2 * half;      // A: VGPR0->K=ka, VGPR1->K=ka+1
    v2f a, bf;
    a.x  = hmat[m * H_ + ka + 0];
    a.y  = hmat[m * H_ + ka + 1];
    // B 4x16: VGPR r holds row K=r (lanes 0-15) / K=r+2 (lanes 16-31)
    bf.x = Wout[n * H_ + ka + 0];
    bf.y = Wout[n * H_ + ka + 1];
    // 8 args: (neg_a, A, neg_b, B, c_mod, C, reuse_a, reuse_b)
    c = __builtin_amdgcn_wmma_f32_16x16x4_f32(
        false, a, false, bf, (short)0, c, false, false);
  }

  // C/D layout: VGPR v, lanes 0-15 -> M=v, N=lane; lanes 16-31 -> M=v+8
  #pragma unroll
  for (int v = 0; v < 8; ++v) {
    const int mg = tileM * 16 + v + 8 * half;
    const int ng = tileN * 16 + l16;
    tag[mg * O_ + ng] = c[v] + bout[ng];
  }
}

// ---------------------------------------------------------------------------
extern "C" void kernel_launch(void* const* d_in, const int* in_sizes, int n_in,
                              void* d_out, int out_size, void* d_ws, size_t ws_size,
                              hipStream_t stream) {
  (void)in_sizes; (void)n_in; (void)out_size; (void)d_ws; (void)ws_size;

  const float* sentence = (const float*)d_in[0];  // [B,T,D]
  const float* h0       = (const float*)d_in[1];  // [B,H]
  const float* F0       = (const float*)d_in[2];  // [B,H,Z]
  const float* W        = (const float*)d_in[3];  // [H,Z]
  const float* bias     = (const float*)d_in[4];  // [H]
  const float* lam      = (const float*)d_in[5];  // [H,Z]
  const float* gam      = (const float*)d_in[6];  // [H,Z]
  const float* Wout     = (const float*)d_in[7];  // [O,H]
  const float* bout     = (const float*)d_in[8];  // [O]

  float* out  = (float*)d_out;
  float* tag  = out;                    // [B,O]
  float* outH = out + B_ * O_;          // [B,H]
  float* outF = outH + B_ * H_;         // [B,H,Z]

  stpn_recurrent_kernel<<<B_, 256, 0, stream>>>(
      sentence, h0, F0, W, bias, lam, gam, outH, outF);

  dim3 grid(B_ / 16, O_ / 16);          // 8 x 8 tiles, one wave each
  head_wmma_kernel<<<grid, 32, 0, stream>>>(outH, Wout, bout, tag);
}